// MessagePassing_31705448579870
// MI455X (gfx1250) — compile-verified
//
#include <hip/hip_runtime.h>

typedef __attribute__((ext_vector_type(2))) float v2f;
typedef __attribute__((ext_vector_type(4))) float v4f;
typedef __attribute__((ext_vector_type(8))) float v8f;
typedef __attribute__((ext_vector_type(4))) int v4i;

#define UNITS 64
#define ADIM 29
#define BDIM 11
#define KTOT 768   // 12*64 (11 bond matrices + bias matrix)
#define GCOLS 192  // 3*UNITS

#if __has_builtin(__builtin_amdgcn_global_load_async_to_lds_b128)
#define HAS_ASYNC_LDS 1
typedef __attribute__((address_space(1))) v4i gas_v4i;   // global
typedef __attribute__((address_space(3))) v4i las_v4i;   // LDS
#else
#define HAS_ASYNC_LDS 0
#endif

__device__ __forceinline__ void async_wait_all() {
#if __has_builtin(__builtin_amdgcn_s_wait_asynccnt)
    __builtin_amdgcn_s_wait_asynccnt(0);
#elif HAS_ASYNC_LDS
    asm volatile("s_wait_asynccnt 0" ::: "memory");
#endif
}

// ---------------- prep kernels ----------------

__global__ void k_pad(const float* __restrict__ af, float* __restrict__ h, int nAtoms) {
    int i = blockIdx.x * blockDim.x + threadIdx.x;
    if (i >= nAtoms * UNITS) return;
    int a = i >> 6, u = i & 63;
    h[i] = (u < ADIM) ? af[a * ADIM + u] : 0.0f;
}

// W2t[n][k], n in 0..63, k = b*64 + j in 0..767
__global__ void k_w2(const float* __restrict__ ek, const float* __restrict__ eb,
                     float* __restrict__ w2t) {
    int i = blockIdx.x * blockDim.x + threadIdx.x;
    if (i >= UNITS * KTOT) return;
    int n = i / KTOT, k = i % KTOT;
    int b = k >> 6, j = k & 63;
    w2t[i] = (b < BDIM) ? ek[b * (UNITS * UNITS) + n * UNITS + j] : eb[n * UNITS + j];
}

// transpose 64x192 -> 192x64
__global__ void k_transpose(const float* __restrict__ src, float* __restrict__ dst) {
    int i = blockIdx.x * blockDim.x + threadIdx.x;
    if (i >= GCOLS * UNITS) return;
    int n = i / UNITS, k = i % UNITS;
    dst[i] = src[k * GCOLS + n];
}

__global__ void k_zero(float* __restrict__ p, int n) {
    int i = blockIdx.x * blockDim.x + threadIdx.x;
    if (i < n) p[i] = 0.0f;
}

// ---------------- edge network: (E x 768) @ (768 x 64) via WMMA f32 16x16x4 ----------------
// block = 128 threads = 4 waves; 64-edge M-tile per block (4 sub-tiles of 16);
// wave w -> output cols [16w,16w+16) for all 4 sub-tiles -> each B fragment feeds 4 WMMAs.
__launch_bounds__(128)
__global__ void k_edge(const float* __restrict__ hIn, const float* __restrict__ bond,
                       const int* __restrict__ pair, const float* __restrict__ w2t,
                       float* __restrict__ msg, int nEdges) {
    __shared__ __align__(16) float sNbr[64][68];
    __shared__ float sCoef[64][12];
    __shared__ int sDst[64];

    const int e0 = blockIdx.x * 64;
    const int tid = threadIdx.x;
    const int lane = tid & 31;
    const int wave = tid >> 5;

    // stage gathered neighbor rows: 64 rows x 64 floats, 16B granules (async copy to LDS)
    for (int r = 0; r < 8; ++r) {
        int idx = tid + r * 128;            // 0..1023 (64 rows x 16 quads)
        int mrow = idx >> 4;
        int u4 = (idx & 15) * 4;
        int e = e0 + mrow;
        if (e < nEdges) {
            int src = pair[e * 2 + 1];
            const float* gp = hIn + (size_t)src * UNITS + u4;
#if HAS_ASYNC_LDS
            __builtin_amdgcn_global_load_async_to_lds_b128(
                (gas_v4i*)(uintptr_t)gp,
                (las_v4i*)(uintptr_t)(unsigned)(uintptr_t)&sNbr[mrow][u4],
                0, 0);
#else
            *(v4f*)&sNbr[mrow][u4] = *(const v4f*)gp;
#endif
        } else {
            v4f z = {0, 0, 0, 0};
            *(v4f*)&sNbr[mrow][u4] = z;
        }
    }
    if (tid < 64) {
        int e = e0 + tid;
        sDst[tid] = (e < nEdges) ? pair[e * 2 + 0] : -1;
    }
    for (int idx = tid; idx < 64 * 12; idx += 128) {
        int mrow = idx / 12, b = idx % 12;
        int e = e0 + mrow;
        sCoef[mrow][b] = (b == BDIM) ? 1.0f : ((e < nEdges) ? bond[e * BDIM + b] : 0.0f);
    }
    async_wait_all();
    __syncthreads();

    const int m = lane & 15;
    const int hh = lane >> 4;               // 0/1: which K-pair this lane holds
    const int n = wave * 16 + (lane & 15);  // output column
    const float* __restrict__ wrow = w2t + (size_t)n * KTOT;

    v8f acc0 = {0, 0, 0, 0, 0, 0, 0, 0};
    v8f acc1 = {0, 0, 0, 0, 0, 0, 0, 0};
    v8f acc2 = {0, 0, 0, 0, 0, 0, 0, 0};
    v8f acc3 = {0, 0, 0, 0, 0, 0, 0, 0};

    for (int b = 0; b < 12; ++b) {
        float cf0 = sCoef[m][b];
        float cf1 = sCoef[m + 16][b];
        float cf2 = sCoef[m + 32][b];
        float cf3 = sCoef[m + 48][b];
#pragma unroll
        for (int kk = 0; kk < 16; ++kk) {
            int kb = b * 64 + kk * 4;
            int c = kk * 4 + 2 * hh;
            v2f b2 = *(const v2f*)&wrow[kb + 2 * hh];   // B: W2t[n][k]
            v2f n0 = *(const v2f*)&sNbr[m][c];
            v2f n1 = *(const v2f*)&sNbr[m + 16][c];
            v2f n2 = *(const v2f*)&sNbr[m + 32][c];
            v2f n3 = *(const v2f*)&sNbr[m + 48][c];
            acc0 = __builtin_amdgcn_wmma_f32_16x16x4_f32(false, n0 * cf0, false, b2, (short)0, acc0, false, false);
            acc1 = __builtin_amdgcn_wmma_f32_16x16x4_f32(false, n1 * cf1, false, b2, (short)0, acc1, false, false);
            acc2 = __builtin_amdgcn_wmma_f32_16x16x4_f32(false, n2 * cf2, false, b2, (short)0, acc2, false, false);
            acc3 = __builtin_amdgcn_wmma_f32_16x16x4_f32(false, n3 * cf3, false, b2, (short)0, acc3, false, false);
        }
    }

    // scatter-add: C element v of sub-tile t -> edge row M = 16t + v + 8*hh, col n
#pragma unroll
    for (int v = 0; v < 8; ++v) {
        int M = v + 8 * hh;
        int d0 = sDst[M];
        int d1 = sDst[M + 16];
        int d2 = sDst[M + 32];
        int d3 = sDst[M + 48];
        if (d0 >= 0) unsafeAtomicAdd(&msg[(size_t)d0 * UNITS + n], acc0[v]);
        if (d1 >= 0) unsafeAtomicAdd(&msg[(size_t)d1 * UNITS + n], acc1[v]);
        if (d2 >= 0) unsafeAtomicAdd(&msg[(size_t)d2 * UNITS + n], acc2[v]);
        if (d3 >= 0) unsafeAtomicAdd(&msg[(size_t)d3 * UNITS + n], acc3[v]);
    }
}

// ---------------- GRU cell: xm = msg@K, hm = h@R via WMMA, gates from LDS ----------------
__launch_bounds__(128)
__global__ void k_gru(const float* __restrict__ msg, const float* __restrict__ hIn,
                      const float* __restrict__ kt, const float* __restrict__ rt,
                      const float* __restrict__ gb, float* __restrict__ hOut, int nAtoms) {
    __shared__ __align__(16) float sX[16][68];
    __shared__ __align__(16) float sH[16][68];
    __shared__ __align__(16) float sXM[16][200];
    __shared__ __align__(16) float sHM[16][200];

    const int a0 = blockIdx.x * 16;
    const int tid = threadIdx.x;
    const int lane = tid & 31;
    const int wave = tid >> 5;

    for (int r = 0; r < 8; ++r) {
        int idx = tid + r * 128;
        int m = idx >> 6, u = idx & 63;
        int a = a0 + m;
        sX[m][u] = (a < nAtoms) ? msg[(size_t)a * UNITS + u] : 0.0f;
        sH[m][u] = (a < nAtoms) ? hIn[(size_t)a * UNITS + u] : 0.0f;
    }
    __syncthreads();

    const int m = lane & 15;
    const int hh = lane >> 4;

#pragma unroll
    for (int t6 = 0; t6 < 6; ++t6) {
        int t = wave * 6 + t6;              // 0..23, uniform per wave
        int isX = (t < 12);
        int tt = isX ? t : t - 12;
        int n = tt * 16 + (lane & 15);
        const float* __restrict__ wrow = (isX ? kt : rt) + (size_t)n * UNITS;
        v8f acc = {0, 0, 0, 0, 0, 0, 0, 0};
        if (isX) {
#pragma unroll
            for (int kk = 0; kk < 16; ++kk) {
                v2f a2 = *(const v2f*)&sX[m][kk * 4 + 2 * hh];
                v2f b2 = *(const v2f*)&wrow[kk * 4 + 2 * hh];
                acc = __builtin_amdgcn_wmma_f32_16x16x4_f32(false, a2, false, b2, (short)0, acc, false, false);
            }
#pragma unroll
            for (int v = 0; v < 8; ++v) sXM[v + 8 * hh][n] = acc[v];
        } else {
#pragma unroll
            for (int kk = 0; kk < 16; ++kk) {
                v2f a2 = *(const v2f*)&sH[m][kk * 4 + 2 * hh];
                v2f b2 = *(const v2f*)&wrow[kk * 4 + 2 * hh];
                acc = __builtin_amdgcn_wmma_f32_16x16x4_f32(false, a2, false, b2, (short)0, acc, false, false);
            }
#pragma unroll
            for (int v = 0; v < 8; ++v) sHM[v + 8 * hh][n] = acc[v];
        }
    }
    __syncthreads();

    for (int r = 0; r < 8; ++r) {
        int idx = tid + r * 128;
        int m2 = idx >> 6, u = idx & 63;
        int a = a0 + m2;
        if (a >= nAtoms) continue;
        float xz = sXM[m2][u]        + gb[u];
        float xr = sXM[m2][u + 64]   + gb[64 + u];
        float xh = sXM[m2][u + 128]  + gb[128 + u];
        float rz = sHM[m2][u]        + gb[GCOLS + u];
        float rr = sHM[m2][u + 64]   + gb[GCOLS + 64 + u];
        float rh = sHM[m2][u + 128]  + gb[GCOLS + 128 + u];
        float z  = 1.0f / (1.0f + __expf(-(xz + rz)));
        float rg = 1.0f / (1.0f + __expf(-(xr + rr)));
        float hv = sH[m2][u];
        float hc = tanhf(xh + rg * rh);
        hOut[(size_t)a * UNITS + u] = z * hv + (1.0f - z) * hc;
    }
}

// ---------------- launcher ----------------

extern "C" void kernel_launch(void* const* d_in, const int* in_sizes, int n_in,
                              void* d_out, int out_size, void* d_ws, size_t ws_size,
                              hipStream_t stream) {
    const float* af   = (const float*)d_in[0];
    const float* bond = (const float*)d_in[1];
    const int*   pair = (const int*)d_in[2];
    const float* ek   = (const float*)d_in[3];
    const float* eb   = (const float*)d_in[4];
    const float* gk   = (const float*)d_in[5];
    const float* gr   = (const float*)d_in[6];
    const float* gb   = (const float*)d_in[7];

    const int nAtoms = in_sizes[0] / ADIM;
    const int nEdges = in_sizes[1] / BDIM;
    const int nh = nAtoms * UNITS;

    char* ws = (char*)d_ws;
    size_t off = 0;
    auto carve = [&](size_t bytes) {
        void* p = ws + off;
        off = (off + bytes + 255) & ~(size_t)255;
        return p;
    };
    float* msg = (float*)carve((size_t)nh * sizeof(float));
    float* hP  = (float*)carve((size_t)nh * sizeof(float));
    float* w2t = (float*)carve((size_t)UNITS * KTOT * sizeof(float));
    float* kt  = (float*)carve((size_t)GCOLS * UNITS * sizeof(float));
    float* rt  = (float*)carve((size_t)GCOLS * UNITS * sizeof(float));
    (void)ws_size; (void)n_in; (void)out_size;

    float* hOutBuf = (float*)d_out;

    k_pad<<<(nh + 255) / 256, 256, 0, stream>>>(af, hOutBuf, nAtoms);
    k_w2<<<(UNITS * KTOT + 255) / 256, 256, 0, stream>>>(ek, eb, w2t);
    k_transpose<<<(GCOLS * UNITS + 255) / 256, 256, 0, stream>>>(gk, kt);
    k_transpose<<<(GCOLS * UNITS + 255) / 256, 256, 0, stream>>>(gr, rt);

    for (int s = 0; s < 4; ++s) {
        float* hin  = (s % 2 == 0) ? hOutBuf : hP;
        float* hout = (s % 2 == 0) ? hP : hOutBuf;   // step 3 lands in d_out
        k_zero<<<(nh + 255) / 256, 256, 0, stream>>>(msg, nh);
        k_edge<<<(nEdges + 63) / 64, 128, 0, stream>>>(hin, bond, pair, w2t, msg, nEdges);
        k_gru<<<(nAtoms + 15) / 16, 128, 0, stream>>>(msg, hin, kt, rt, gb, hout, nAtoms);
    }
}